// RNNModel_35579509080499
// MI455X (gfx1250) — compile-verified
//
#include <hip/hip_runtime.h>

typedef __attribute__((ext_vector_type(2))) float v2f;
typedef __attribute__((ext_vector_type(4))) float v4f;
typedef __attribute__((ext_vector_type(8))) float v8f;

constexpr int kBatch  = 1024;
constexpr int kSeq    = 512;
constexpr int kInDim  = 100;
constexpr int kHid    = 10;
constexpr int kPrePad = 12;   // pre rows padded to 12 floats (48B, 16B-aligned)

// ---------------------------------------------------------------------------
// Kernel 1: embedding gather + input projection (pre = emb[x] @ W_ih^T + b_ih)
// One wave per 16x16 tile: M = 16 batch rows (fixed s), N = 16 (hidden padded
// from 10), K = 100 input dim, via 25 x V_WMMA_F32_16X16X4_F32.
// ---------------------------------------------------------------------------
__global__ __launch_bounds__(256) void rnn_pre_kernel(
    const int* __restrict__ x, const float* __restrict__ emb,
    const float* __restrict__ W_ih, const float* __restrict__ b_ih,
    float* __restrict__ pre)
{
    // B-matrix source: Wpad[n][k] = W_ih[n][k] for n<10, zero for n in [10,16)
    __shared__ float Wpad[16 * kInDim];
    for (int idx = threadIdx.x; idx < 16 * kInDim; idx += blockDim.x) {
        int n = idx / kInDim, k = idx % kInDim;
        Wpad[idx] = (n < kHid) ? W_ih[n * kInDim + k] : 0.0f;
    }
    __syncthreads();

    const int lane = threadIdx.x & 31;
    const int half = lane >> 4;        // 0: holds K {0,1}; 1: holds K {2,3}
    const int mrow = lane & 15;        // A: M index, B/D: N index
    const int wid  = blockIdx.x * (blockDim.x >> 5) + (threadIdx.x >> 5);
    const int s    = wid >> 6;         // sequence position   [0, 512)
    const int b0   = (wid & 63) << 4;  // batch tile start    {0,16,...,1008}

    // Gather: lane supplies A[M=mrow][K=kk+2*half .. +1] = emb[tok(b0+mrow)][...]
    const int  b   = b0 + mrow;
    const int  tok = x[b * kSeq + s];
    const float* __restrict__ arow = emb  + (long)tok * kInDim + 2 * half;
    const float* __restrict__ brow = Wpad + mrow * kInDim      + 2 * half;

    v8f c = {};
    #pragma unroll
    for (int kk = 0; kk < kInDim; kk += 4) {
        // A 16x4 f32 layout: v0 = A[m][kk + 2*half], v1 = A[m][kk + 2*half + 1]
        v2f a  = *(const v2f*)(arow + kk);
        // B 4x16 f32 layout (mirror of A): v0 = B[kk + 2*half][n] = W_ih[n][kk + 2*half]
        v2f bf = *(const v2f*)(brow + kk);
        c = __builtin_amdgcn_wmma_f32_16x16x4_f32(
                /*neg_a=*/false, a, /*neg_b=*/false, bf,
                /*c_mod=*/(short)0, c, /*reuse_a=*/false, /*reuse_b=*/false);
    }

    // D layout: lane holds D[M = v + 8*half][N = mrow] for v = 0..7
    const int n = mrow;
    if (n < kPrePad) {
        const float bias = (n < kHid) ? b_ih[n] : 0.0f;
        #pragma unroll
        for (int v = 0; v < 8; ++v) {
            const int m = v + 8 * half;
            pre[((long)s * kBatch + b0 + m) * kPrePad + n] = c[v] + bias;
        }
    }
}

// ---------------------------------------------------------------------------
// Kernel 2: sequential RNN scan + fc + sigmoid. One lane per batch row.
// W_hh/b_hh/W_fc/b_fc are wave-uniform -> scalar loads / SGPR-resident.
// ---------------------------------------------------------------------------
__global__ __launch_bounds__(256) void rnn_scan_kernel(
    const float* __restrict__ pre, const float* __restrict__ W_hh,
    const float* __restrict__ b_hh, const float* __restrict__ W_fc,
    const float* __restrict__ b_fc, float* __restrict__ out)
{
    const int b = blockIdx.x * blockDim.x + threadIdx.x;
    if (b >= kBatch) return;

    float w[kHid][kHid], bb[kHid];
    #pragma unroll
    for (int i = 0; i < kHid; ++i) {
        bb[i] = b_hh[i];
        #pragma unroll
        for (int j = 0; j < kHid; ++j) w[i][j] = W_hh[i * kHid + j];
    }

    float h[kHid];
    #pragma unroll
    for (int i = 0; i < kHid; ++i) h[i] = 0.0f;

    const float* __restrict__ p = pre + (long)b * kPrePad;
    for (int t = 0; t < kSeq; ++t) {
        const v4f p0 = *(const v4f*)(p + 0);
        const v4f p1 = *(const v4f*)(p + 4);
        const v4f p2 = *(const v4f*)(p + 8);
        const float pt[kHid] = {p0.x, p0.y, p0.z, p0.w,
                                p1.x, p1.y, p1.z, p1.w,
                                p2.x, p2.y};
        float nh[kHid];
        #pragma unroll
        for (int i = 0; i < kHid; ++i) {
            float acc = pt[i] + bb[i];
            #pragma unroll
            for (int j = 0; j < kHid; ++j) acc = fmaf(h[j], w[i][j], acc);
            nh[i] = fmaxf(acc, 0.0f);
        }
        #pragma unroll
        for (int i = 0; i < kHid; ++i) h[i] = nh[i];
        p += (long)kBatch * kPrePad;
    }

    float o = b_fc[0];
    #pragma unroll
    for (int i = 0; i < kHid; ++i) o = fmaf(h[i], W_fc[i], o);
    out[b] = 1.0f / (1.0f + expf(-o));
}

// ---------------------------------------------------------------------------
extern "C" void kernel_launch(void* const* d_in, const int* in_sizes, int n_in,
                              void* d_out, int out_size, void* d_ws, size_t ws_size,
                              hipStream_t stream) {
    (void)in_sizes; (void)n_in; (void)out_size; (void)ws_size;
    const int*   x    = (const int*)  d_in[0];
    const float* emb  = (const float*)d_in[1];
    const float* W_ih = (const float*)d_in[2];
    const float* b_ih = (const float*)d_in[3];
    const float* W_hh = (const float*)d_in[4];
    const float* b_hh = (const float*)d_in[5];
    const float* W_fc = (const float*)d_in[6];
    const float* b_fc = (const float*)d_in[7];
    float* out = (float*)d_out;
    float* pre = (float*)d_ws;   // needs 512*1024*12*4 = ~25.2 MB

    // 64 batch tiles * 512 seq positions = 32768 waves; 8 waves per block.
    rnn_pre_kernel<<<4096, 256, 0, stream>>>(x, emb, W_ih, b_ih, pre);
    rnn_scan_kernel<<<(kBatch + 255) / 256, 256, 0, stream>>>(
        pre, W_hh, b_hh, W_fc, b_fc, out);
}